// DiscreteLoss_74655121539644
// MI455X (gfx1250) — compile-verified
//
#include <hip/hip_runtime.h>
#include <hip/hip_bf16.h>
#include <math.h>

typedef __attribute__((ext_vector_type(2))) float v2f;
typedef __attribute__((ext_vector_type(8))) float v8f;

#define NB 512
#define NT 256

// Problem dims (fixed by the reference setup)
#define DIM_B 64
#define DIM_S 128
#define DIM_N 128
#define DIM_D 512
#define DIM_V 128

// Wave-level sum via V_WMMA_F32_16X16X4_F32:
// A (16x4 f32): lane m supplies {w,0} as K=0,1 ; lane m+16 supplies {w,0} as K=2,3
// B = ones (4x16)  =>  D[m,n] = A-row-sum = w[m] + w[m+16]  for every n.
// C/D layout: lane L holds D[j + 8*(L>>4), L&15] in c[j]  =>
// sum_j c[j] over lane L plus lane L^16 = total wave sum.
__device__ __forceinline__ float wave_sum_wmma(float w) {
    v2f a; a.x = w;    a.y = 0.0f;
    v2f b; b.x = 1.0f; b.y = 1.0f;
    v8f c = {};
    c = __builtin_amdgcn_wmma_f32_16x16x4_f32(
        /*neg_a=*/false, a, /*neg_b=*/false, b,
        /*c_mod=*/(short)0, c, /*reuse_a=*/false, /*reuse_b=*/false);
    float s = c[0] + c[1] + c[2] + c[3] + c[4] + c[5] + c[6] + c[7];
    s += __shfl_xor(s, 16, 32);
    return s;  // every lane holds the wave total
}

__device__ __forceinline__ float sq4(float4 a, float4 b, float acc) {
    float dx = a.x - b.x, dy = a.y - b.y, dz = a.z - b.z, dw = a.w - b.w;
    acc = fmaf(dx, dx, acc);
    acc = fmaf(dy, dy, acc);
    acc = fmaf(dz, dz, acc);
    acc = fmaf(dw, dw, acc);
    return acc;
}

__global__ __launch_bounds__(NT) void loss_partial_kernel(
    const float* __restrict__ zs, const float* __restrict__ pts,
    const float* __restrict__ masks, const float* __restrict__ rzs,
    const float* __restrict__ qy, const float* __restrict__ logits,
    const float* __restrict__ best_pt, const float* __restrict__ best_mask,
    const float* __restrict__ best_rz, const float* __restrict__ pts_gt,
    const float* __restrict__ masks_gt, const float* __restrict__ best_pt_gt,
    const float* __restrict__ best_mask_gt, const long long* __restrict__ mapping,
    float* __restrict__ partials)
{
    const int tid    = blockIdx.x * blockDim.x + threadIdx.x;
    const int stride = gridDim.x * blockDim.x;
    const int mark[4] = {0, 29, 88, 117};

    const float4* __restrict__ zs4       = (const float4*)zs;
    const float4* __restrict__ rzs4      = (const float4*)rzs;
    const float4* __restrict__ pts4      = (const float4*)pts;
    const float4* __restrict__ pts_gt4   = (const float4*)pts_gt;
    const float4* __restrict__ masks4    = (const float4*)masks;
    const float4* __restrict__ masks_gt4 = (const float4*)masks_gt;
    const float4* __restrict__ qy4       = (const float4*)qy;

    float a_auto = 0.f, a_seg = 0.f, a_disk = 0.f, a_kld = 0.f, a_lm = 0.f;
    float a_bauto = 0.f, a_bseg = 0.f, a_bdisk = 0.f, a_blm = 0.f;

    // ---- auto: sum_d (rzs[b, map[b,s], d] - zs[b,s,d])^2, vectorized x4 (D=512) ----
    {
        const int D4 = DIM_D / 4;
        for (int i = tid; i < DIM_B * DIM_S * D4; i += stride) {
            int d4 = i % D4;
            int bs = i / D4;
            int b  = bs / DIM_S;
            int m  = (int)mapping[bs];
            float4 r = rzs4[(size_t)(b * DIM_S + m) * D4 + d4];
            float4 z = zs4[i];
            a_auto = sq4(r, z, a_auto);
        }
    }
    // ---- seg + disk share index space (B,S,N*2) and the same gather, vectorized x4 ----
    {
        const int R4 = (DIM_N * 2) / 4;  // 64 float4 per (b,s) row
        for (int i = tid; i < DIM_B * DIM_S * R4; i += stride) {
            int nc4 = i % R4;
            int bs  = i / R4;
            int b   = bs / DIM_S;
            int m   = (int)mapping[bs];
            size_t gi = (size_t)(b * DIM_S + m) * R4 + nc4;
            a_seg  = sq4(masks4[gi], masks_gt4[i], a_seg);
            a_disk = sq4(pts4[gi],   pts_gt4[i],   a_disk);
        }
    }
    // ---- kld: qy * (log(qy+eps) + log(V)), vectorized x4 ----
    {
        const float logV = logf((float)DIM_V);
        for (int i = tid; i < (DIM_B * DIM_S * DIM_V) / 4; i += stride) {
            float4 q = qy4[i];
            a_kld = fmaf(q.x, logf(q.x + 1e-20f) + logV, a_kld);
            a_kld = fmaf(q.y, logf(q.y + 1e-20f) + logV, a_kld);
            a_kld = fmaf(q.z, logf(q.z + 1e-20f) + logV, a_kld);
            a_kld = fmaf(q.w, logf(q.w + 1e-20f) + logV, a_kld);
        }
    }
    // ---- landmark (gathered pts at 4 mark columns) ----
    for (int i = tid; i < DIM_B * DIM_S * 4 * 2; i += stride) {
        int c  = i & 1;
        int k  = (i >> 1) & 3;
        int bs = i >> 3;
        int b  = bs / DIM_S;
        int m  = (int)mapping[bs];
        int n  = mark[k];
        float d = pts[(((size_t)(b * DIM_S + m)) * DIM_N + n) * 2 + c]
                - pts_gt[(((size_t)bs) * DIM_N + n) * 2 + c];
        a_lm = fmaf(d, d, a_lm);
    }
    // ---- best_auto: (best_rz - logits)^2 over (B,D) ----
    for (int i = tid; i < DIM_B * DIM_D; i += stride) {
        float d = best_rz[i] - logits[i];
        a_bauto = fmaf(d, d, a_bauto);
    }
    // ---- best_seg: N-slice [32:96] of best_mask ----
    for (int i = tid; i < DIM_B * 64 * 2; i += stride) {
        int c = i & 1;
        int n = 32 + ((i >> 1) & 63);
        int b = i >> 7;
        size_t idx = ((size_t)b * DIM_N + n) * 2 + c;
        float d = best_mask[idx] - best_mask_gt[idx];
        a_bseg = fmaf(d, d, a_bseg);
    }
    // ---- best_disk over (B,N,2) ----
    for (int i = tid; i < DIM_B * DIM_N * 2; i += stride) {
        float d = best_pt[i] - best_pt_gt[i];
        a_bdisk = fmaf(d, d, a_bdisk);
    }
    // ---- best_landmark ----
    for (int i = tid; i < DIM_B * 4 * 2; i += stride) {
        int c = i & 1;
        int k = (i >> 1) & 3;
        int b = i >> 3;
        size_t idx = ((size_t)b * DIM_N + mark[k]) * 2 + c;
        float d = best_pt[idx] - best_pt_gt[idx];
        a_blm = fmaf(d, d, a_blm);
    }

    // Fold normalizations (ALPHA=1, BETA=0.1, GAMMA=1, final *B) into weights.
    const double w_auto  = (double)DIM_B / DIM_D;                               // 0.125
    const double w_disk  = 1.0 / DIM_S;
    const double w_lm    = 1.0 / DIM_S;
    const double w_seg   = 1.0 / ((double)DIM_S * DIM_N * 2.0);
    const double w_kld   = 0.1 / DIM_S;
    const double w_bauto = 1.0 / DIM_D;
    const double w_bseg  = 1.0 / 128.0;
    const double w_bdisk = 1.0 / ((double)DIM_N * 2.0 * DIM_B * DIM_N);
    const double w_blm   = 1.0 / DIM_N;

    double tacc = w_auto * a_auto + w_seg * a_seg + w_disk * a_disk
                + w_kld * a_kld + w_lm * a_lm + w_bauto * a_bauto
                + w_bseg * a_bseg + w_bdisk * a_bdisk + w_blm * a_blm;

    // Wave reduction via WMMA (EXEC all ones: full block, no divergence above)
    float wsum = wave_sum_wmma((float)tacc);

    __shared__ float lds[NT / 32];
    int lane = threadIdx.x & 31;
    int wave = threadIdx.x >> 5;
    if (lane == 0) lds[wave] = wsum;
    __syncthreads();
    if (threadIdx.x == 0) {
        float bsum = 0.f;
        for (int w = 0; w < NT / 32; ++w) bsum += lds[w];
        partials[blockIdx.x] = bsum;
    }
}

__global__ __launch_bounds__(32) void loss_final_kernel(
    const float* __restrict__ partials, int nb, float* __restrict__ out)
{
    double acc = 0.0;
    for (int i = threadIdx.x; i < nb; i += 32) acc += (double)partials[i];
    float tot = wave_sum_wmma((float)acc);
    if (threadIdx.x == 0) out[0] = tot;
}

extern "C" void kernel_launch(void* const* d_in, const int* in_sizes, int n_in,
                              void* d_out, int out_size, void* d_ws, size_t ws_size,
                              hipStream_t stream) {
    const float*     zs           = (const float*)d_in[0];
    const float*     pts          = (const float*)d_in[1];
    const float*     masks        = (const float*)d_in[2];
    const float*     rzs          = (const float*)d_in[3];
    const float*     qy           = (const float*)d_in[4];
    const float*     logits       = (const float*)d_in[5];
    const float*     best_pt      = (const float*)d_in[6];
    const float*     best_mask    = (const float*)d_in[7];
    const float*     best_rz      = (const float*)d_in[8];
    const float*     pts_gt       = (const float*)d_in[9];
    const float*     masks_gt     = (const float*)d_in[10];
    const float*     best_pt_gt   = (const float*)d_in[11];
    const float*     best_mask_gt = (const float*)d_in[12];
    const long long* mapping      = (const long long*)d_in[13];
    // d_in[14] = vector_dims scalar (=128), baked into the kernel constants.

    float* partials = (float*)d_ws;  // NB floats = 2 KB of scratch

    loss_partial_kernel<<<NB, NT, 0, stream>>>(
        zs, pts, masks, rzs, qy, logits, best_pt, best_mask, best_rz,
        pts_gt, masks_gt, best_pt_gt, best_mask_gt, mapping, partials);

    loss_final_kernel<<<1, 32, 0, stream>>>(partials, NB, (float*)d_out);
}